// MyModel_5609227289087
// MI455X (gfx1250) — compile-verified
//
#include <hip/hip_runtime.h>
#include <math.h>

#define SEQ   2048
#define DIM   1280
#define NHEAD 16
#define HD    80
#define QKVN  (3*DIM)

typedef __attribute__((ext_vector_type(2))) float v2f;
typedef __attribute__((ext_vector_type(8))) float v8f;

// D = A(16x4) * B(4x16) + C, fp32 WMMA (wave32)
__device__ __forceinline__ v8f wmma4(v2f a, v2f b, v8f c) {
  return __builtin_amdgcn_wmma_f32_16x16x4_f32(false, a, false, b, (short)0, c,
                                               false, false);
}

// ---------------------------------------------------------------------------
// Kernel 1: qkv = H @ W_qkv + b_qkv, scattered into q/k/v buffers [h][s][d]
// One wave per 16x16 output tile. 2048/16 x 3840/16 = 128 x 240 tiles.
// ---------------------------------------------------------------------------
__global__ __launch_bounds__(256) void qkv_gemm_kernel(
    const float* __restrict__ H, const float* __restrict__ W,
    const float* __restrict__ bias,
    float* __restrict__ qbuf, float* __restrict__ kbuf, float* __restrict__ vbuf)
{
  const int lane = threadIdx.x & 31;
  const int wave = threadIdx.x >> 5;
  const int tile = blockIdx.x * 8 + wave;
  const int tm = tile / (QKVN / 16);
  const int tn = tile % (QKVN / 16);
  const int m0 = tm * 16, n0 = tn * 16;
  const int mA = lane & 15;            // A row / B,C column within tile
  const int kOff = (lane >> 4) * 2;    // K sub-offset per A/B layout

  v8f acc = {};
  const float* Arow = H + (size_t)(m0 + mA) * DIM;
  for (int kk = 0; kk < DIM; kk += 4) {
    v2f a = *(const v2f*)(Arow + kk + kOff);           // A: m=lane&15, k=v+kOff
    v2f b;
    b.x = W[(size_t)(kk + kOff + 0) * QKVN + n0 + mA]; // B: k=v+kOff, n=lane&15
    b.y = W[(size_t)(kk + kOff + 1) * QKVN + n0 + mA];
    acc = wmma4(a, b, acc);
  }

  // Epilogue: bias + scatter to q/k/v in [head][seq][dim] layout.
  const int n = n0 + mA;               // global column in [0,3840)
  const float bn = bias[n];
  const int which = n / DIM;
  const int dd = n % DIM;
  const int h = dd / HD;
  const int d = dd % HD;
  float* dst = (which == 0) ? qbuf : (which == 1) ? kbuf : vbuf;
  dst += (size_t)h * SEQ * HD + d;
  const int mBase = m0 + (lane >> 4) * 8; // C: row = r + 8*(lane>>4)
  #pragma unroll
  for (int r = 0; r < 8; ++r)
    dst[(size_t)(mBase + r) * HD] = acc[r] + bn;
}

// ---------------------------------------------------------------------------
// Kernel 2: rotary embedding applied in place to q and k.
// One thread per (d, d+40) pair to avoid in-place races.
// ---------------------------------------------------------------------------
__global__ void rope_kernel(float* __restrict__ q, float* __restrict__ k,
                            const float* __restrict__ rope)
{
  const int tid = blockIdx.x * blockDim.x + threadIdx.x;
  if (tid >= NHEAD * SEQ * (HD / 2)) return;
  const int d = tid % (HD / 2);
  const int s = (tid / (HD / 2)) % SEQ;
  const int h = tid / (SEQ * (HD / 2));
  const float f = rope[s * (HD / 2) + d];
  const float c = cosf(f), sn = sinf(f);
  const size_t i0 = (size_t)h * SEQ * HD + (size_t)s * HD + d;
  const size_t i1 = i0 + HD / 2;
  const float q0 = q[i0], q1 = q[i1];
  q[i0] = q0 * c - q1 * sn;
  q[i1] = q1 * c + q0 * sn;
  const float k0 = k[i0], k1 = k[i1];
  k[i0] = k0 * c - k1 * sn;
  k[i1] = k1 * c + k0 * sn;
}

// ---------------------------------------------------------------------------
// Kernel 3: flash attention. One wave per (head, 16-query block).
// Streams 128 key tiles with online softmax; P converted C->A layout via LDS.
// ---------------------------------------------------------------------------
__global__ __launch_bounds__(256) void attn_kernel(
    const float* __restrict__ qbuf, const float* __restrict__ kbuf,
    const float* __restrict__ vbuf, const float* __restrict__ mask,
    float* __restrict__ ctx)
{
  __shared__ float ldsP[8][16 * 20];   // per-wave 16x16 P tile, row stride 20
  const int lane = threadIdx.x & 31;
  const int wave = threadIdx.x >> 5;
  const int wid  = blockIdx.x * 8 + wave;
  const int h  = wid / (SEQ / 16);
  const int qb = wid % (SEQ / 16);
  const int q0 = qb * 16;
  const int mA = lane & 15;
  const int hi = lane >> 4;
  const int kOff = hi * 2;
  const float scale = rsqrtf((float)HD);

  const float* Q = qbuf + (size_t)h * SEQ * HD;
  const float* K = kbuf + (size_t)h * SEQ * HD;
  const float* V = vbuf + (size_t)h * SEQ * HD;

  // Hoist Q tile in A-operand layout: 20 K-blocks of 4.
  v2f qa[20];
  #pragma unroll
  for (int t = 0; t < 20; ++t)
    qa[t] = *(const v2f*)(Q + (size_t)(q0 + mA) * HD + t * 4 + kOff);

  v8f o0 = {}, o1 = {}, o2 = {}, o3 = {}, o4 = {};
  float rmax[8], rsum[8];
  #pragma unroll
  for (int r = 0; r < 8; ++r) { rmax[r] = -INFINITY; rsum[r] = 0.f; }

  float* P = ldsP[wave];

  for (int kb = 0; kb < SEQ / 16; ++kb) {
    const int k0 = kb * 16;

    // S = Q * K^T (16x16), K-dim = 80 in 20 WMMA steps.
    v8f s = {};
    #pragma unroll
    for (int t = 0; t < 20; ++t) {
      v2f bK = *(const v2f*)(K + (size_t)(k0 + mA) * HD + t * 4 + kOff);
      s = wmma4(qa[t], bK, s);
    }

    // Online softmax. Row r of the C tile lives in VGPR r across a 16-lane
    // group (lanes 0-15: rows 0-7, lanes 16-31: rows 8-15).
    #pragma unroll
    for (int r = 0; r < 8; ++r) {
      float x = s[r] * scale +
                mask[(size_t)(q0 + r + 8 * hi) * SEQ + k0 + mA];
      float tm = x;
      tm = fmaxf(tm, __shfl_xor(tm, 1, 32));
      tm = fmaxf(tm, __shfl_xor(tm, 2, 32));
      tm = fmaxf(tm, __shfl_xor(tm, 4, 32));
      tm = fmaxf(tm, __shfl_xor(tm, 8, 32));
      const float mnew  = fmaxf(rmax[r], tm);
      const float alpha = __expf(rmax[r] - mnew);
      const float p     = __expf(x - mnew);
      float ts = p;
      ts += __shfl_xor(ts, 1, 32);
      ts += __shfl_xor(ts, 2, 32);
      ts += __shfl_xor(ts, 4, 32);
      ts += __shfl_xor(ts, 8, 32);
      rsum[r] = rsum[r] * alpha + ts;
      rmax[r] = mnew;
      o0[r] *= alpha; o1[r] *= alpha; o2[r] *= alpha; o3[r] *= alpha; o4[r] *= alpha;
      P[(r + 8 * hi) * 20 + mA] = p;   // C-layout -> plain row-major in LDS
    }

    // Reload P in A-operand layout (same-wave DS ops are in order).
    v2f pa[4];
    #pragma unroll
    for (int j = 0; j < 4; ++j)
      pa[j] = *(const v2f*)(P + mA * 20 + j * 4 + kOff);

    // O += P * V  (16x16 x 16x80 -> five 16x16 tiles).
    #pragma unroll
    for (int j = 0; j < 4; ++j) {
      const float* Vr = V + (size_t)(k0 + j * 4 + kOff) * HD + mA;
      v2f b;
      b.x = Vr[0];   b.y = Vr[HD];       o0 = wmma4(pa[j], b, o0);
      b.x = Vr[16];  b.y = Vr[HD + 16];  o1 = wmma4(pa[j], b, o1);
      b.x = Vr[32];  b.y = Vr[HD + 32];  o2 = wmma4(pa[j], b, o2);
      b.x = Vr[48];  b.y = Vr[HD + 48];  o3 = wmma4(pa[j], b, o3);
      b.x = Vr[64];  b.y = Vr[HD + 64];  o4 = wmma4(pa[j], b, o4);
    }
  }

  // Normalize and store into ctx [seq][DIM] (head-interleaved columns).
  #pragma unroll
  for (int r = 0; r < 8; ++r) {
    const float inv = 1.0f / rsum[r];
    const size_t row = (size_t)(q0 + r + 8 * hi) * DIM + h * HD + mA;
    ctx[row +  0] = o0[r] * inv;
    ctx[row + 16] = o1[r] * inv;
    ctx[row + 32] = o2[r] * inv;
    ctx[row + 48] = o3[r] * inv;
    ctx[row + 64] = o4[r] * inv;
  }
}

// ---------------------------------------------------------------------------
// Kernel 4: out = ctx @ W_proj + b_proj. 128 x 80 tiles.
// ---------------------------------------------------------------------------
__global__ __launch_bounds__(256) void proj_gemm_kernel(
    const float* __restrict__ A, const float* __restrict__ W,
    const float* __restrict__ bias, float* __restrict__ out)
{
  const int lane = threadIdx.x & 31;
  const int wave = threadIdx.x >> 5;
  const int tile = blockIdx.x * 8 + wave;
  const int tm = tile / (DIM / 16);
  const int tn = tile % (DIM / 16);
  const int m0 = tm * 16, n0 = tn * 16;
  const int mA = lane & 15;
  const int kOff = (lane >> 4) * 2;

  v8f acc = {};
  const float* Arow = A + (size_t)(m0 + mA) * DIM;
  for (int kk = 0; kk < DIM; kk += 4) {
    v2f a = *(const v2f*)(Arow + kk + kOff);
    v2f b;
    b.x = W[(size_t)(kk + kOff + 0) * DIM + n0 + mA];
    b.y = W[(size_t)(kk + kOff + 1) * DIM + n0 + mA];
    acc = wmma4(a, b, acc);
  }
  const float bn = bias[n0 + mA];
  const int mBase = m0 + (lane >> 4) * 8;
  #pragma unroll
  for (int r = 0; r < 8; ++r)
    out[(size_t)(mBase + r) * DIM + n0 + mA] = acc[r] + bn;
}

// ---------------------------------------------------------------------------
extern "C" void kernel_launch(void* const* d_in, const int* in_sizes, int n_in,
                              void* d_out, int out_size, void* d_ws, size_t ws_size,
                              hipStream_t stream)
{
  const float* H     = (const float*)d_in[0];  // hidden_states [2048,1280]
  const float* mask  = (const float*)d_in[1];  // attention_mask [1,2048,2048]
  const float* rope  = (const float*)d_in[2];  // rotary_pos_emb [2048,40]
  const float* Wqkv  = (const float*)d_in[3];  // [1280,3840]
  const float* bqkv  = (const float*)d_in[4];  // [3840]
  const float* Wproj = (const float*)d_in[5];  // [1280,1280]
  const float* bproj = (const float*)d_in[6];  // [1280]
  float* out = (float*)d_out;                  // [2048,1280]

  const size_t perBuf = (size_t)NHEAD * SEQ * HD; // 2.62M floats each
  float* qbuf = (float*)d_ws;
  float* kbuf = qbuf + perBuf;
  float* vbuf = kbuf + perBuf;
  float* ctx  = vbuf + perBuf;                    // [2048,1280]

  qkv_gemm_kernel<<<(SEQ / 16) * (QKVN / 16) / 8, 256, 0, stream>>>(
      H, Wqkv, bqkv, qbuf, kbuf, vbuf);
  rope_kernel<<<(NHEAD * SEQ * (HD / 2)) / 256, 256, 0, stream>>>(
      qbuf, kbuf, rope);
  attn_kernel<<<(NHEAD * (SEQ / 16)) / 8, 256, 0, stream>>>(
      qbuf, kbuf, vbuf, mask, ctx);
  proj_gemm_kernel<<<(SEQ / 16) * (DIM / 16) / 8, 256, 0, stream>>>(
      ctx, Wproj, bproj, out);
}